// SparseArch_41566693490909
// MI455X (gfx1250) — compile-verified
//
#include <hip/hip_runtime.h>

// SparseArch embedding-bag pooling for MI455X (gfx1250, wave32).
// HBM-bound gather: ~406 MB of traffic -> ~17us floor at 23.3 TB/s.
// CDNA5 async global->LDS DMA (ASYNCcnt) with double buffering; all bag-level
// control data forced wave-uniform (SALU/SMEM) via readfirstlane.

#define B_BATCH   16384
#define DIM       128
#define NFEAT     4
#define WPB       8                       // waves per block (wave32)
#define BLOCK     (WPB * 32)              // 256 threads
#define NPAIR     (B_BATCH * NFEAT)       // 65536 (bag, feature) pairs
#define NBLK      (NPAIR / WPB)           // 8192 blocks
#define PRED_COLS (NFEAT * DIM)           // 512
#define PRED_ELEMS ((long long)B_BATCH * PRED_COLS)

// Exact parameter type of __builtin_amdgcn_global_load_async_to_lds_b128
// (from clang diagnostic): pointer to 16-byte int vector, AS1 / AS3.
typedef int vint4 __attribute__((vector_size(16)));
typedef __attribute__((address_space(1))) float as1_float;
typedef __attribute__((address_space(3))) float as3_float;
typedef __attribute__((address_space(1))) vint4 as1_vint4;
typedef __attribute__((address_space(3))) vint4 as3_vint4;

// ---- CDNA5 async global->LDS copy: 16 bytes per lane ------------------------
__device__ __forceinline__ void async_copy16(const float* src, float* lds_dst) {
#if defined(__HIP_DEVICE_COMPILE__)
#if __has_builtin(__builtin_amdgcn_global_load_async_to_lds_b128)
  as1_vint4* g = (as1_vint4*)(as1_float*)src;
  as3_vint4* l = (as3_vint4*)(as3_float*)lds_dst;
  __builtin_amdgcn_global_load_async_to_lds_b128(g, l, 0, 0);
#else
  unsigned loff = (unsigned)(unsigned long long)(as3_float*)lds_dst;
  asm volatile("global_load_async_to_lds_b128 %0, %1, off"
               :: "v"(loff), "v"(src)
               : "memory");
#endif
#endif
}

template <int N>
__device__ __forceinline__ void wait_async() {
#if defined(__HIP_DEVICE_COMPILE__)
#if __has_builtin(__builtin_amdgcn_s_wait_asynccnt)
  __builtin_amdgcn_s_wait_asynccnt(N);
#else
  if (N == 0) asm volatile("s_wait_asynccnt 0" ::: "memory");
  else        asm volatile("s_wait_asynccnt 1" ::: "memory");
#endif
  asm volatile("" ::: "memory");   // compiler barrier: LDS reads stay after wait
#endif
}

__device__ __forceinline__ int uniform(int v) {
#if defined(__HIP_DEVICE_COMPILE__)
  return __builtin_amdgcn_readfirstlane(v);
#else
  return v;
#endif
}

// ---- Stage 1: pooled embeddings + per-block partial sums --------------------
__global__ __launch_bounds__(BLOCK) void sparse_pool_kernel(
    const int* __restrict__ idx0, const int* __restrict__ len0,
    const float* __restrict__ emb0, const float* __restrict__ pw0,
    const int* __restrict__ idx1, const int* __restrict__ len1,
    const float* __restrict__ emb1, const float* __restrict__ pw1,
    const int* __restrict__ idx2, const int* __restrict__ len2,
    const float* __restrict__ emb2, const float* __restrict__ pw2,
    const int* __restrict__ idx3, const int* __restrict__ len3,
    const float* __restrict__ emb3, const float* __restrict__ pw3,
    float* __restrict__ pred, float* __restrict__ block_sums)
{
  __shared__ __align__(16) float rows[WPB][2][DIM];  // double buffer per wave
  __shared__ float wsum[WPB];

  const int lane = threadIdx.x & 31;
  const int warp = uniform(threadIdx.x >> 5);        // wave-uniform in SGPR
  const int pair = uniform(blockIdx.x * WPB + warp); // < NPAIR, uniform
  const int bag  = pair >> 2;
  const int f    = pair & 3;

  // Scalar-friendly per-feature selection (no memory const tables).
  const int    Lf  = (f >= 2) ? 12 : 10;
  const int*   idx_tab[4] = {idx0, idx1, idx2, idx3};
  const int*   len_tab[4] = {len0, len1, len2, len3};
  const float* emb_tab[4] = {emb0, emb1, emb2, emb3};
  const float* pw_tab[4]  = {pw0, pw1, pw2, pw3};

  const float* emb = emb_tab[f];
  const float* pw  = pw_tab[f];
  int len = len_tab[f][bag];                 // uniform address -> s_load
  len = len < 0 ? 0 : (len > Lf ? Lf : len);
  len = uniform(len);
  const int* bagIdx = idx_tab[f] + bag * Lf;

  float ax = 0.f, ay = 0.f, az = 0.f, aw = 0.f;

  if (len > 0) {
    {
      const long long r0 = bagIdx[0];
      async_copy16(emb + r0 * DIM + lane * 4, &rows[warp][0][lane * 4]);
    }
    for (int l = 0; l < len; ++l) {
      if (l + 1 < len) {
        const long long rn = bagIdx[l + 1];
        async_copy16(emb + rn * DIM + lane * 4, &rows[warp][(l + 1) & 1][lane * 4]);
        wait_async<1>();     // oldest (row l) has landed; next is in flight
      } else {
        wait_async<0>();
      }
      const float4 v = *(const float4*)(&rows[warp][l & 1][lane * 4]);
      const float wl = pw[l];
      ax = fmaf(wl, v.x, ax);
      ay = fmaf(wl, v.y, ay);
      az = fmaf(wl, v.z, az);
      aw = fmaf(wl, v.w, aw);
    }
  }

  // Coalesced pred store: wave writes one contiguous 512B feature slice.
  float4 out4;
  out4.x = ax; out4.y = ay; out4.z = az; out4.w = aw;
  *(float4*)(pred + (size_t)bag * PRED_COLS + (size_t)f * DIM + lane * 4) = out4;

  // Deterministic loss partials: wave32 shuffle reduce -> block sum.
  float s = ax + ay + az + aw;
#pragma unroll
  for (int off = 16; off > 0; off >>= 1)
    s += __shfl_xor(s, off, 32);
  if (lane == 0) wsum[warp] = s;
  __syncthreads();
  if (threadIdx.x == 0) {
    float t = 0.f;
#pragma unroll
    for (int w = 0; w < WPB; ++w) t += wsum[w];
    block_sums[blockIdx.x] = t;
  }
}

// ---- Stage 2: deterministic final reduction -> loss -------------------------
__global__ __launch_bounds__(256) void reduce_loss_kernel(
    const float* __restrict__ block_sums, float* __restrict__ loss_out)
{
  __shared__ float sh[256];
  float s = 0.f;
  for (int i = threadIdx.x; i < NBLK; i += 256) s += block_sums[i];
  sh[threadIdx.x] = s;
  __syncthreads();
  for (int off = 128; off > 0; off >>= 1) {
    if (threadIdx.x < off) sh[threadIdx.x] += sh[threadIdx.x + off];
    __syncthreads();
  }
  if (threadIdx.x == 0)
    loss_out[0] = sh[0] * (1.0f / (float)PRED_ELEMS);
}

extern "C" void kernel_launch(void* const* d_in, const int* in_sizes, int n_in,
                              void* d_out, int out_size, void* d_ws, size_t ws_size,
                              hipStream_t stream) {
  // setup_inputs() dict order: per feature f: idx{f}, len{f}, emb{f}, pw{f}
  const int*   idx[4]; const int* len[4]; const float* emb[4]; const float* pw[4];
  for (int f = 0; f < 4; ++f) {
    idx[f] = (const int*)  d_in[4 * f + 0];
    len[f] = (const int*)  d_in[4 * f + 1];
    emb[f] = (const float*)d_in[4 * f + 2];
    pw[f]  = (const float*)d_in[4 * f + 3];
  }
  float* out        = (float*)d_out;
  float* loss       = out;        // d_out[0]
  float* pred       = out + 1;    // [B, 512] row-major
  float* block_sums = (float*)d_ws;  // NBLK floats (32 KB)

  sparse_pool_kernel<<<NBLK, BLOCK, 0, stream>>>(
      idx[0], len[0], emb[0], pw[0],
      idx[1], len[1], emb[1], pw[1],
      idx[2], len[2], emb[2], pw[2],
      idx[3], len[3], emb[3], pw[3],
      pred, block_sums);

  reduce_loss_kernel<<<1, 256, 0, stream>>>(block_sums, loss);
}